// ImageDiffeomorphismOLD_8143257993686
// MI455X (gfx1250) — compile-verified
//
#include <hip/hip_runtime.h>

typedef __attribute__((ext_vector_type(2))) float v2f;
typedef __attribute__((ext_vector_type(8))) float v8f;

#define N_IMG   224
#define M_MODES 32
#define NPIX    (224 * 224)   // 50176
#define NPLANES 384           // 128 * 3
#define SCALE_DISP 22.4f      // sqrt(T=0.01) * n=224  -> 0.1 * 224

// ---------------------------------------------------------------------------
// ws layout (float units unless noted):
//   S    : [224*32]  @ float off 0        (7168 floats)
//   Cu   : [32*32]   @ float off 7168     (1024)
//   Cv   : [32*32]   @ float off 8192     (1024)
//   M1u  : [224*32]  @ float off 9216     (7168)
//   M1v  : [224*32]  @ float off 16384    (7168)   -> ends at 23552 floats = 94208 B
//   idx  : int2 [NPIX]   @ byte off 94208   (401408 B)
//   w    : float2[NPIX]  @ byte off 495616  (401408 B)   total = 897024 B
// ---------------------------------------------------------------------------

// Kernel 0: sine basis S[x][i] = sin(pi * (x/223) * (i+1)), and masked
// coefficients C = F * e, e = (r < 32.5)/r, r = sqrt((i+1)^2+(j+1)^2).
__global__ void prep_kernel(const float* __restrict__ Fx, const float* __restrict__ Fy,
                            float* __restrict__ S, float* __restrict__ Cu,
                            float* __restrict__ Cv) {
    const int t = blockIdx.x * blockDim.x + threadIdx.x;   // 0..9215
    const float PI = 3.14159265358979323846f;
    if (t < N_IMG * M_MODES) {
        int x = t >> 5;
        int i = t & 31;
        float xs = (float)x * (1.0f / 223.0f);
        S[t] = sinf(PI * xs * (float)(i + 1));
    } else {
        int t2  = t - N_IMG * M_MODES;                     // 0..2047
        int fld = t2 >> 10;                                // 0 = u(Fx), 1 = v(Fy)
        int ij  = t2 & 1023;
        int i   = ij >> 5;
        int j   = ij & 31;
        float ki = (float)(i + 1), kj = (float)(j + 1);
        float r  = sqrtf(ki * ki + kj * kj);
        float e  = (r < 32.5f) ? (1.0f / r) : 0.0f;
        if (fld == 0) Cu[ij] = Fx[ij] * e;
        else          Cv[ij] = Fy[ij] * e;
    }
}

// f32 WMMA tile layouts (wave32, ISA 7.12.2):
//  A 16x4 : lanes 0-15 -> M=lane, vgpr0=K, vgpr1=K+1 ; lanes 16-31 -> K+2,K+3
//  B 4x16 : lanes give N; vgpr0 = rows K / K+2 (lane halves), vgpr1 = K+1 / K+3
//  C/D    : vgpr r -> M = r (+8 for upper lane half), N = lane%16

// Kernel 1: M1 = S @ C  (224x32 = 224x32 @ 32x32), one wave per 16x16 tile.
__global__ void gemm1_kernel(const float* __restrict__ S, const float* __restrict__ Cu,
                             const float* __restrict__ Cv, float* __restrict__ M1u,
                             float* __restrict__ M1v) {
    const int lane = threadIdx.x;
    const int half = lane >> 4;       // 0 or 1
    const int ln   = lane & 15;
    const int m0   = blockIdx.x * 16; // 14 tiles over M=224
    const int n0   = blockIdx.y * 16; // 2 tiles over N=32
    const float* C  = blockIdx.z ? Cv  : Cu;
    float*       M1 = blockIdx.z ? M1v : M1u;

    v8f acc = {0.f, 0.f, 0.f, 0.f, 0.f, 0.f, 0.f, 0.f};
    for (int k = 0; k < M_MODES; k += 4) {
        const int ka = k + 2 * half;
        v2f a, b;
        a.x = S[(m0 + ln) * M_MODES + ka];
        a.y = S[(m0 + ln) * M_MODES + ka + 1];
        b.x = C[ka * M_MODES + n0 + ln];
        b.y = C[(ka + 1) * M_MODES + n0 + ln];
        acc = __builtin_amdgcn_wmma_f32_16x16x4_f32(false, a, false, b,
                                                    (short)0, acc, false, false);
    }
    for (int r = 0; r < 8; ++r)
        M1[(m0 + r + 8 * half) * M_MODES + n0 + ln] = acc[r];
}

// Kernel 2: field = S @ M1^T for BOTH fields (shared A operand), then fuse the
// clip/floor into per-pixel bilinear precomp: idx=(xf,yf), w=(xv,yv).
__global__ void gemm2_kernel(const float* __restrict__ S, const float* __restrict__ M1u,
                             const float* __restrict__ M1v, int2* __restrict__ idx,
                             float2* __restrict__ w) {
    const int lane = threadIdx.x;
    const int half = lane >> 4;
    const int ln   = lane & 15;
    const int m0   = blockIdx.x * 16;   // y tile (M)
    const int n0   = blockIdx.y * 16;   // x tile (N)

    v8f accU = {0.f, 0.f, 0.f, 0.f, 0.f, 0.f, 0.f, 0.f};
    v8f accV = {0.f, 0.f, 0.f, 0.f, 0.f, 0.f, 0.f, 0.f};
    for (int k = 0; k < M_MODES; k += 4) {
        const int ka = k + 2 * half;
        v2f a, bu, bv;
        a.x  = S[(m0 + ln) * M_MODES + ka];
        a.y  = S[(m0 + ln) * M_MODES + ka + 1];
        // B[K][N] = M1^T[k+K][n0+N] = M1[n0+N][k+K]
        bu.x = M1u[(n0 + ln) * M_MODES + ka];
        bu.y = M1u[(n0 + ln) * M_MODES + ka + 1];
        bv.x = M1v[(n0 + ln) * M_MODES + ka];
        bv.y = M1v[(n0 + ln) * M_MODES + ka + 1];
        accU = __builtin_amdgcn_wmma_f32_16x16x4_f32(false, a, false, bu,
                                                     (short)0, accU, false, false);
        accV = __builtin_amdgcn_wmma_f32_16x16x4_f32(false, a, false, bv,
                                                     (short)0, accV, false, false);
    }
    for (int r = 0; r < 8; ++r) {
        const int y = m0 + r + 8 * half;
        const int x = n0 + ln;
        float dx = SCALE_DISP * accU[r];
        float dy = SCALE_DISP * accV[r];
        float xn = fminf(fmaxf((float)x + dx, 0.0f), 223.0f);
        float yn = fminf(fmaxf((float)y + dy, 0.0f), 223.0f);
        float xff = floorf(xn);
        float yff = floorf(yn);
        const int p = y * N_IMG + x;
        idx[p] = make_int2((int)xff, (int)yff);
        w[p]   = make_float2(xn - xff, yn - yff);
    }
}

// Kernel 3: bilinear gather remap. One thread per output pixel; precomp is
// ~800KB and stays L2-resident across all 384 plane replays; source planes
// (77MB total) fit in MI455X's 192MB L2; output stored non-temporal so it
// doesn't displace the gather working set.
__global__ void remap_kernel(const float* __restrict__ x, const int2* __restrict__ idx,
                             const float2* __restrict__ w, float* __restrict__ out) {
    const int p     = blockIdx.x * blockDim.x + threadIdx.x;  // 0..50175
    const int plane = blockIdx.y;                             // 0..383
    const int2   ij = idx[p];
    const float2 wv = w[p];
    const int xf = ij.x, yf = ij.y;
    const int xc = min(xf + 1, N_IMG - 1);   // == ceil(xn) whenever its weight != 0
    const int yc = min(yf + 1, N_IMG - 1);
    const float* a = x + (size_t)plane * NPIX;
    const float v00 = a[yf * N_IMG + xf];
    const float v01 = a[yf * N_IMG + xc];
    const float v10 = a[yc * N_IMG + xf];
    const float v11 = a[yc * N_IMG + xc];
    const float xv = wv.x, yv = wv.y;
    const float omx = 1.0f - xv, omy = 1.0f - yv;
    const float res = omy * omx * v00 + omy * xv * v01
                    + yv  * omx * v10 + yv  * xv * v11;
    __builtin_nontemporal_store(res, out + (size_t)plane * NPIX + p);
}

extern "C" void kernel_launch(void* const* d_in, const int* in_sizes, int n_in,
                              void* d_out, int out_size, void* d_ws, size_t ws_size,
                              hipStream_t stream) {
    const float* x  = (const float*)d_in[0];
    const float* Fx = (const float*)d_in[1];
    const float* Fy = (const float*)d_in[2];
    float* out = (float*)d_out;

    float* ws   = (float*)d_ws;
    float* S    = ws;
    float* Cu   = ws + 7168;
    float* Cv   = ws + 8192;
    float* M1u  = ws + 9216;
    float* M1v  = ws + 16384;
    int2*   idx = (int2*)((char*)d_ws + 94208);
    float2* wgt = (float2*)((char*)d_ws + 495616);

    // 36*256 = 9216 = 7168 (S) + 2048 (Cu,Cv)
    prep_kernel<<<36, 256, 0, stream>>>(Fx, Fy, S, Cu, Cv);
    gemm1_kernel<<<dim3(14, 2, 2), 32, 0, stream>>>(S, Cu, Cv, M1u, M1v);
    gemm2_kernel<<<dim3(14, 14), 32, 0, stream>>>(S, M1u, M1v, idx, wgt);
    remap_kernel<<<dim3(NPIX / 256, NPLANES), 256, 0, stream>>>(x, idx, wgt, out);
}